// Net_32779190403593
// MI455X (gfx1250) — compile-verified
//
#include <hip/hip_runtime.h>

#define N_NODES   524288
#define N_EDGES   16777216
#define NODE_ATOM 64
#define N_H1      1024
#define DIM_OUT   128
#define N_ROWS    (N_NODES / NODE_ATOM)   // 8192

// Fused-MLP tiling
#define MLP_ROWS  32                      // rows per block
#define S1        1028                    // LDS pitch for H tile (== 4 mod 64: conflict-free b64 frag reads)
#define SG        68                      // LDS pitch for G tile (== 4 mod 64)
#define LDS_BYTES ((MLP_ROWS * S1 + MLP_ROWS * SG) * 4)   // 140,288 B (2 blocks/WGP fit in 320 KB)

typedef __attribute__((ext_vector_type(2))) float v2f;
typedef __attribute__((ext_vector_type(8))) float v8f;

// ---------------------------------------------------------------------------
// Zero the three scatter accumulators (must run every launch: harness does not
// re-initialize the workspace between graph replays).
// ---------------------------------------------------------------------------
__global__ void zero3_kernel(float* __restrict__ a, float* __restrict__ b,
                             float* __restrict__ c) {
    int i = blockIdx.x * blockDim.x + threadIdx.x;
    a[i] = 0.0f;
    b[i] = 0.0f;
    c[i] = 0.0f;
}

// ---------------------------------------------------------------------------
// Edge pass 1: gated message (sigmoid * softplus) scatter-added to dst,
// plus weighted degree accumulation. Memory-bound: 20 B/edge streaming
// (int64 index pair + edge_attr) + L2-resident x gathers + L2 float atomics.
// ---------------------------------------------------------------------------
__global__ void edge_pass1(const float* __restrict__ x,
                           const float* __restrict__ ea,
                           const long long* __restrict__ ei,
                           const float* __restrict__ wf, const float* __restrict__ bf,
                           const float* __restrict__ wsv, const float* __restrict__ bs,
                           float* __restrict__ hsum, float* __restrict__ deg) {
    int e = blockIdx.x * blockDim.x + threadIdx.x;
    int src = (int)ei[e];
    int dst = (int)ei[N_EDGES + e];
    float xs = x[src];
    float xd = x[dst];
    float w  = ea[e];
    // z = [x[dst], x[src], edge_attr]
    float tf = xd * wf[0] + xs * wf[1] + w * wf[2] + bf[0];
    float ts = xd * wsv[0] + xs * wsv[1] + w * wsv[2] + bs[0];
    float sig = 1.0f / (1.0f + __expf(-tf));
    float sp  = fmaxf(ts, 0.0f) + log1pf(__expf(-fabsf(ts)));  // stable softplus
    atomicAdd(&hsum[dst], sig * sp);
    atomicAdd(&deg[dst], w);
}

// ---------------------------------------------------------------------------
// Node pass 1: h = relu(x + hsum), dinv = deg>0 ? rsqrt(max(deg,1e-12)) : 0
// ---------------------------------------------------------------------------
__global__ void node_pass1(const float* __restrict__ x,
                           const float* __restrict__ hsum,
                           const float* __restrict__ deg,
                           float* __restrict__ hrel,
                           float* __restrict__ dinv) {
    int i = blockIdx.x * blockDim.x + threadIdx.x;
    hrel[i] = fmaxf(x[i] + hsum[i], 0.0f);
    float d = deg[i];
    dinv[i] = (d > 0.0f) ? rsqrtf(fmaxf(d, 1e-12f)) : 0.0f;
}

// ---------------------------------------------------------------------------
// Edge pass 2 (GCN): acc2[dst] += dinv[src]*w*dinv[dst] * h[src]*gcn_w
// ---------------------------------------------------------------------------
__global__ void edge_pass2(const float* __restrict__ ea,
                           const long long* __restrict__ ei,
                           const float* __restrict__ dinv,
                           const float* __restrict__ hrel,
                           const float* __restrict__ gw,
                           float* __restrict__ acc2) {
    int e = blockIdx.x * blockDim.x + threadIdx.x;
    int src = (int)ei[e];
    int dst = (int)ei[N_EDGES + e];
    float w = ea[e];
    float v = dinv[src] * w * dinv[dst] * hrel[src] * gw[0];
    atomicAdd(&acc2[dst], v);
}

// ---------------------------------------------------------------------------
// Node pass 2: g = relu(acc2 + gcn_b)   (g viewed as 8192 x 64 row-major)
// ---------------------------------------------------------------------------
__global__ void node_pass2(const float* __restrict__ acc2,
                           const float* __restrict__ gb,
                           float* __restrict__ gmat) {
    int i = blockIdx.x * blockDim.x + threadIdx.x;
    gmat[i] = fmaxf(acc2[i] + gb[0], 0.0f);
}

// ---------------------------------------------------------------------------
// Fused MLP:  out = relu( relu(bn(G @ W3^T + b3)) @ W4^T + b4 )
//
// One block = 32 rows of G. 8 waves / 256 threads.
//  Phase 0: stage G tile (32x64) -> LDS (pitch 68).
//  Phase 1: each wave owns (mtile = wave>>2, ntile-group = wave&3) and emits
//           16 H tiles of 16x16 via V_WMMA_F32_16X16X4_F32 (A from LDS,
//           B = W3 rows from global/L2), fusing bias + BN + ReLU, storing the
//           hidden activations to the 32x1024 LDS tile (pitch 1028).
//  Phase 2: each wave owns two adjacent 16x16 output tiles sharing one
//           A-fragment stream: 256 k-steps, 1 ds_load_b64 feeds 2 WMMAs.
//
// This removes the 64 MB H write+read entirely (the whole hidden tile lives
// in the WGP's 320 KB LDS) and makes all fragment reads LDS-bank-conflict
// free (pitches == 4 mod 64 spread a wave's b64 reads over all 64 banks).
// Per-lane fragment layout for 16x16x4 f32 (ISA 7.12.2): lanes 0-15 hold
// K={k0,k0+1}, lanes 16-31 hold K={k0+2,k0+3}; A lane = row M, B lane = col N.
// ---------------------------------------------------------------------------
__global__ void mlp_fused_wmma(const float* __restrict__ G,     // 8192 x 64
                               const float* __restrict__ W3,    // 1024 x 64
                               const float* __restrict__ b3,
                               const float* __restrict__ gamma,
                               const float* __restrict__ beta,
                               const float* __restrict__ W4,    // 128 x 1024
                               const float* __restrict__ b4,
                               float* __restrict__ out) {       // 8192 x 128
    extern __shared__ float smem[];
    float* Hs = smem;                       // MLP_ROWS x S1
    float* Gs = smem + MLP_ROWS * S1;       // MLP_ROWS x SG

    const int tid  = threadIdx.x;
    const int lane = tid & 31;
    const int wave = tid >> 5;              // 0..7
    const int half = lane >> 4;
    const int l    = lane & 15;
    const int rowbase = blockIdx.x * MLP_ROWS;

    // ---- Phase 0: stage G tile (rows are contiguous: 32*64 floats) ----
    for (int i = tid; i < MLP_ROWS * NODE_ATOM; i += 256) {
        int r = i >> 6;
        int c = i & 63;
        Gs[r * SG + c] = G[rowbase * NODE_ATOM + i];
    }
    __syncthreads();

    // ---- Phase 1: H tile = relu(bn(G @ W3^T + b3)) ----
    {
        const int mt  = wave >> 2;           // 0..1
        const int ntg = wave & 3;            // 0..3
        const float* Arl = Gs + (mt * 16 + l) * SG;     // LDS A row for lane
        const float bnscale = 0.99999500003749973f;     // 1/sqrt(1 + 1e-5)
        for (int j = 0; j < 16; ++j) {
            const int nt = ntg * 16 + j;                // 0..63
            const float* Br = W3 + (nt * 16 + l) * NODE_ATOM;
            v8f c = {};
#pragma unroll
            for (int k0 = 0; k0 < NODE_ATOM; k0 += 4) {
                const int kb = k0 + 2 * half;           // even -> 8B aligned
                v2f a = *(const v2f*)(Arl + kb);        // ds_load_b64
                v2f b = *(const v2f*)(Br + kb);         // global_load_b64
                c = __builtin_amdgcn_wmma_f32_16x16x4_f32(false, a, false, b,
                                                          (short)0, c, false, false);
            }
            const int n = nt * 16 + l;
            const float bias = b3[n];
            const float ga = gamma[n] * bnscale;
            const float be = beta[n];
#pragma unroll
            for (int r = 0; r < 8; ++r) {
                const int row = mt * 16 + r + 8 * half;
                Hs[row * S1 + n] = fmaxf(ga * (c[r] + bias) + be, 0.0f);
            }
        }
    }
    __syncthreads();

    // ---- Phase 2: out tile = relu(H @ W4^T + b4) ----
    {
        const int t0  = wave * 2;            // two adjacent tiles share mt
        const int mt  = t0 >> 3;             // 0..1
        const int nt0 = t0 & 7;              // nt0, nt0+1
        const float* Arl = Hs + (mt * 16 + l) * S1;
        const float* Br0 = W4 + ((nt0 + 0) * 16 + l) * N_H1;
        const float* Br1 = W4 + ((nt0 + 1) * 16 + l) * N_H1;
        v8f c0 = {};
        v8f c1 = {};
#pragma unroll 4
        for (int k0 = 0; k0 < N_H1; k0 += 4) {
            const int kb = k0 + 2 * half;
            v2f a  = *(const v2f*)(Arl + kb);           // one LDS read ...
            v2f bb0 = *(const v2f*)(Br0 + kb);
            v2f bb1 = *(const v2f*)(Br1 + kb);
            c0 = __builtin_amdgcn_wmma_f32_16x16x4_f32(false, a, false, bb0,
                                                       (short)0, c0, false, false);
            c1 = __builtin_amdgcn_wmma_f32_16x16x4_f32(false, a, false, bb1,
                                                       (short)0, c1, false, false);
        }
        const int n0 = (nt0 + 0) * 16 + l;
        const int n1 = (nt0 + 1) * 16 + l;
        const float bias0 = b4[n0];
        const float bias1 = b4[n1];
#pragma unroll
        for (int r = 0; r < 8; ++r) {
            const int row = rowbase + mt * 16 + r + 8 * half;
            out[row * DIM_OUT + n0] = fmaxf(c0[r] + bias0, 0.0f);
            out[row * DIM_OUT + n1] = fmaxf(c1[r] + bias1, 0.0f);
        }
    }
}

// ---------------------------------------------------------------------------
// Launch
// ---------------------------------------------------------------------------
extern "C" void kernel_launch(void* const* d_in, const int* in_sizes, int n_in,
                              void* d_out, int out_size, void* d_ws, size_t ws_size,
                              hipStream_t stream) {
    const float*     x     = (const float*)d_in[0];
    const float*     ea    = (const float*)d_in[1];
    const float*     wf    = (const float*)d_in[2];
    const float*     bf    = (const float*)d_in[3];
    const float*     wsv   = (const float*)d_in[4];
    const float*     bs    = (const float*)d_in[5];
    const float*     gw    = (const float*)d_in[6];
    const float*     gb    = (const float*)d_in[7];
    const float*     W3    = (const float*)d_in[8];
    const float*     b3    = (const float*)d_in[9];
    const float*     gamma = (const float*)d_in[10];
    const float*     beta  = (const float*)d_in[11];
    const float*     W4    = (const float*)d_in[12];
    const float*     b4    = (const float*)d_in[13];
    const long long* ei    = (const long long*)d_in[14];
    float*           out   = (float*)d_out;

    // Workspace carve-up (floats): 6 * N_NODES = 12 MB
    float* ws   = (float*)d_ws;
    float* hsum = ws + 0 * (size_t)N_NODES;
    float* deg  = ws + 1 * (size_t)N_NODES;
    float* hrel = ws + 2 * (size_t)N_NODES;
    float* dinv = ws + 3 * (size_t)N_NODES;
    float* acc2 = ws + 4 * (size_t)N_NODES;
    float* gmat = ws + 5 * (size_t)N_NODES;

    // Allow >48KB dynamic LDS for the fused MLP kernel (CDNA5: 320 KB / WGP).
    (void)hipFuncSetAttribute((const void*)mlp_fused_wmma,
                              hipFuncAttributeMaxDynamicSharedMemorySize,
                              LDS_BYTES);

    zero3_kernel<<<N_NODES / 256, 256, 0, stream>>>(hsum, deg, acc2);
    edge_pass1 <<<N_EDGES / 256, 256, 0, stream>>>(x, ea, ei, wf, bf, wsv, bs, hsum, deg);
    node_pass1 <<<N_NODES / 256, 256, 0, stream>>>(x, hsum, deg, hrel, dinv);
    edge_pass2 <<<N_EDGES / 256, 256, 0, stream>>>(ea, ei, dinv, hrel, gw, acc2);
    node_pass2 <<<N_NODES / 256, 256, 0, stream>>>(acc2, gb, gmat);

    mlp_fused_wmma<<<N_ROWS / MLP_ROWS, 256, LDS_BYTES, stream>>>(
        gmat, W3, b3, gamma, beta, W4, b4, out);
}